// ScaledDotProductAttention_2_3624952398646
// MI455X (gfx1250) — compile-verified
//
#include <hip/hip_runtime.h>

typedef __attribute__((ext_vector_type(2))) float v2f;
typedef __attribute__((ext_vector_type(8))) float v8f;

#define BH   64          // B*H
#define SEQ  512         // L
#define DH   64          // head dim
#define QK_SCALE 0.125f  // 1/temperature (temp = 8)
#define EPS_ 1.0f
#define LOG_MU_ (-6.238324625039508f)   // -log(512)

__device__ __forceinline__ float wave_max(float x) {
#pragma unroll
  for (int off = 16; off >= 1; off >>= 1)
    x = fmaxf(x, __shfl_xor(x, off, 32));
  return x;
}
__device__ __forceinline__ float wave_sum(float x) {
#pragma unroll
  for (int off = 16; off >= 1; off >>= 1)
    x += __shfl_xor(x, off, 32);
  return x;
}

// ---------------------------------------------------------------------------
// C[bh,i,j] = dot(q[bh,i,:], k[bh,j,:]) / 8     via V_WMMA_F32_16X16X4_F32
// One 16x16 tile per wave; 8 waves per 256-thread block.
// A fragment: lane%16 = M row of q-tile, (lane/16)*2 = K pair within chunk.
// B fragment: lane%16 = N col (= k row), (lane/16)*2 = K pair (k is row-major,
// so B loads are contiguous float2 from k's rows).
// ---------------------------------------------------------------------------
__global__ __launch_bounds__(256) void qk_gemm_kernel(
    const float* __restrict__ q, const float* __restrict__ k,
    float* __restrict__ Cmat) {
  const int lane = threadIdx.x & 31;
  const int wave = threadIdx.x >> 5;
  const int tile = blockIdx.x * 8 + wave;     // 0 .. 65535
  const int bh = tile >> 10;
  const int t  = tile & 1023;
  const int ti = t >> 5;                      // 0..31  (row tile)
  const int tj = t & 31;                      // 0..31  (col tile)
  const int halfk = (lane >> 4) * 2;          // 0 or 2

  const float* qrow = q + ((size_t)bh * SEQ + ti * 16 + (lane & 15)) * DH;
  const float* krow = k + ((size_t)bh * SEQ + tj * 16 + (lane & 15)) * DH;

  v8f acc = {0.f, 0.f, 0.f, 0.f, 0.f, 0.f, 0.f, 0.f};
#pragma unroll
  for (int d = 0; d < DH; d += 4) {
    const float2 qa = *(const float2*)(qrow + d + halfk);
    const float2 kb = *(const float2*)(krow + d + halfk);
    v2f a; a.x = qa.x * QK_SCALE; a.y = qa.y * QK_SCALE;
    v2f b; b.x = kb.x; b.y = kb.y;
    acc = __builtin_amdgcn_wmma_f32_16x16x4_f32(false, a, false, b,
                                                (short)0, acc, false, false);
  }

  // C/D layout: VGPR r, lanes 0-15 -> row r, lanes 16-31 -> row r+8; col = lane%16
  float* out = Cmat + (size_t)bh * SEQ * SEQ + (size_t)(ti * 16) * SEQ +
               tj * 16 + (lane & 15);
  const int rbase = (lane >> 4) * 8;
#pragma unroll
  for (int r = 0; r < 8; ++r)
    out[(size_t)(r + rbase) * SEQ] = acc[r];
}

// ---------------------------------------------------------------------------
// u, v dual potentials init to zero
// ---------------------------------------------------------------------------
__global__ __launch_bounds__(256) void init_uv_kernel(float* __restrict__ buf) {
  int idx = blockIdx.x * 256 + threadIdx.x;   // 65536 = 2 * BH * SEQ
  if (idx < 2 * BH * SEQ) buf[idx] = 0.f;
}

// ---------------------------------------------------------------------------
// u-update: u_i = eps*(log_mu - LSE_j((-C_ij + u_i + v_j)/eps)) + u_i
// One wave per row (coalesced 128B reads), register-held row slice.
// ---------------------------------------------------------------------------
__global__ __launch_bounds__(256) void sinkhorn_u_kernel(
    const float* __restrict__ Cmat, float* __restrict__ u,
    const float* __restrict__ vdual) {
  const int lane = threadIdx.x & 31;
  const int row = blockIdx.x * 8 + (threadIdx.x >> 5);   // 0..32767
  const int bh = row >> 9;
  const float* Crow = Cmat + (size_t)row * SEQ;
  const float* vrow = vdual + bh * SEQ;
  const float u_i = u[row];

  float x[16];
  float m = -3.4e38f;
#pragma unroll
  for (int t = 0; t < 16; ++t) {
    const int j = lane + 32 * t;
    x[t] = (u_i + vrow[j] - Crow[j]) * (1.0f / EPS_);
    m = fmaxf(m, x[t]);
  }
  m = wave_max(m);
  float s = 0.f;
#pragma unroll
  for (int t = 0; t < 16; ++t) s += expf(x[t] - m);
  s = wave_sum(s);
  if (lane == 0) {
    const float lse = m + logf(s);
    u[row] = EPS_ * (LOG_MU_ - lse) + u_i;
  }
}

// ---------------------------------------------------------------------------
// v-update: v_j = eps*(log_nu - LSE_i((-C_ij + u_i + v_j)/eps)) + v_j
// One thread per column; sweep rows with coalesced 64-wide loads and a
// branchless online logsumexp (single pass over C).
// ---------------------------------------------------------------------------
__global__ __launch_bounds__(64) void sinkhorn_v_kernel(
    const float* __restrict__ Cmat, const float* __restrict__ u,
    float* __restrict__ vdual) {
  const int bh = blockIdx.x >> 3;
  const int j = ((blockIdx.x & 7) << 6) + threadIdx.x;
  const float* Cbase = Cmat + (size_t)bh * SEQ * SEQ + j;
  const float* ubh = u + bh * SEQ;
  const float v_j = vdual[bh * SEQ + j];

  float m = -3.4e38f, s = 0.f;
  for (int i = 0; i < SEQ; ++i) {
    const float xv = (ubh[i] + v_j - Cbase[(size_t)i * SEQ]) * (1.0f / EPS_);
    const float mn = fmaxf(m, xv);
    s = s * expf(m - mn) + expf(xv - mn);
    m = mn;
  }
  const float lse = m + logf(s);
  vdual[bh * SEQ + j] = EPS_ * (LOG_MU_ - lse) + v_j;
}

// ---------------------------------------------------------------------------
// In-place: attnw = exp((-C + u_i + v_j)/eps) * L     (float4 vectorized)
// ---------------------------------------------------------------------------
__global__ __launch_bounds__(256) void finalize_kernel(
    float* __restrict__ attnw, const float* __restrict__ u,
    const float* __restrict__ vdual) {
  const size_t quad = (size_t)blockIdx.x * 256 + threadIdx.x;  // 4,194,304 total
  const size_t base = quad * 4;
  const int bh = (int)(base >> 18);
  const int i = (int)((base >> 9) & (SEQ - 1));
  const int j0 = (int)(base & (SEQ - 1));
  const float uu = u[bh * SEQ + i];
  const float* vb = vdual + bh * SEQ + j0;
  float4 c = *(float4*)(attnw + base);
  float4 r;
  r.x = expf((uu + vb[0] - c.x) * (1.0f / EPS_)) * (float)SEQ;
  r.y = expf((uu + vb[1] - c.y) * (1.0f / EPS_)) * (float)SEQ;
  r.z = expf((uu + vb[2] - c.z) * (1.0f / EPS_)) * (float)SEQ;
  r.w = expf((uu + vb[3] - c.w) * (1.0f / EPS_)) * (float)SEQ;
  *(float4*)(attnw + base) = r;
}

// ---------------------------------------------------------------------------
// out[bh,i,d] = sum_j attnw[bh,i,j] * v[bh,j,d]   via fp32 WMMA, K=512
// ---------------------------------------------------------------------------
__global__ __launch_bounds__(256) void pv_gemm_kernel(
    const float* __restrict__ attnw, const float* __restrict__ vmat,
    float* __restrict__ out) {
  const int lane = threadIdx.x & 31;
  const int wave = threadIdx.x >> 5;
  const int tile = blockIdx.x * 8 + wave;   // 0..8191
  const int bh = tile >> 7;                 // 128 tiles per bh
  const int t = tile & 127;
  const int ti = t >> 2;                    // 0..31 (rows)
  const int td = t & 3;                     // 0..3  (cols of D=64)
  const int halfk = (lane >> 4) * 2;

  const float* Arow = attnw + ((size_t)bh * SEQ + ti * 16 + (lane & 15)) * SEQ;
  const float* Bcol = vmat + (size_t)bh * SEQ * DH + td * 16 + (lane & 15);

  v8f acc = {0.f, 0.f, 0.f, 0.f, 0.f, 0.f, 0.f, 0.f};
#pragma unroll 8
  for (int j = 0; j < SEQ; j += 4) {
    const float2 af = *(const float2*)(Arow + j + halfk);
    v2f a; a.x = af.x; a.y = af.y;
    v2f b;
    b.x = Bcol[(size_t)(j + halfk) * DH];
    b.y = Bcol[(size_t)(j + halfk + 1) * DH];
    acc = __builtin_amdgcn_wmma_f32_16x16x4_f32(false, a, false, b,
                                                (short)0, acc, false, false);
  }

  float* obase = out + ((size_t)bh * SEQ + ti * 16) * DH + td * 16 + (lane & 15);
  const int rbase = (lane >> 4) * 8;
#pragma unroll
  for (int r = 0; r < 8; ++r)
    obase[(size_t)(r + rbase) * DH] = acc[r];
}

// ---------------------------------------------------------------------------
extern "C" void kernel_launch(void* const* d_in, const int* in_sizes, int n_in,
                              void* d_out, int out_size, void* d_ws,
                              size_t ws_size, hipStream_t stream) {
  (void)in_sizes; (void)n_in; (void)out_size; (void)ws_size;
  const float* q = (const float*)d_in[0];
  const float* k = (const float*)d_in[1];
  const float* v = (const float*)d_in[2];

  float* out = (float*)d_out;                        // [64,512,64]
  float* attnw = out + (size_t)BH * SEQ * DH;        // [64,512,512] (C, then pi*L)
  float* u = (float*)d_ws;                           // [64,512]
  float* vdual = u + BH * SEQ;                       // [64,512]

  // 1) cost matrices C = (q/temp) @ k^T  (stored in d_out attn region)
  qk_gemm_kernel<<<8192, 256, 0, stream>>>(q, k, attnw);
  // 2) zero dual potentials
  init_uv_kernel<<<256, 256, 0, stream>>>(u);
  // 3) 10 Sinkhorn iterations (u then v, matching the reference scan body)
  for (int it = 0; it < 10; ++it) {
    sinkhorn_u_kernel<<<4096, 256, 0, stream>>>(attnw, u, vdual);
    sinkhorn_v_kernel<<<512, 64, 0, stream>>>(attnw, u, vdual);
  }
  // 4) attn_weights = exp(M(u,v)) * L   (in place over C)
  finalize_kernel<<<16384, 256, 0, stream>>>(attnw, u, vdual);
  // 5) output = attn_weights @ v
  pv_gemm_kernel<<<1024, 256, 0, stream>>>(attnw, v, out);
}